// WeightedAverage_color_55551107006569
// MI455X (gfx1250) — compile-verified
//
#include <hip/hip_runtime.h>
#include <hip/hip_bf16.h>
#include <stdint.h>

// ---------------------------------------------------------------------------
// WeightedAverage_color for MI455X (gfx1250).
// Fused single pass: rgb->pca, 7x7 softmax-weighted smoothing, pca->rgb.
// CDNA5 paths: async global->LDS halo staging (ASYNCcnt) + s_wait_asynccnt,
// and V_WMMA_F32_16X16X4_F32 for the pca->rgb 3x3 transform epilogue.
// ---------------------------------------------------------------------------

typedef __attribute__((ext_vector_type(2))) float v2f;
typedef __attribute__((ext_vector_type(8))) float v8f;

#define AS1 __attribute__((address_space(1)))
#define AS3 __attribute__((address_space(3)))

#if __has_builtin(__builtin_amdgcn_global_load_async_to_lds_b32)
#define HAVE_ASYNC 1
#endif
#if __has_builtin(__builtin_amdgcn_wmma_f32_16x16x4_f32)
#define HAVE_WMMA 1
#endif

namespace {
constexpr int Wimg  = 256;
constexpr int Himg  = 256;
constexpr int PLANE = Wimg * Himg;          // 65536
constexpr int TILE  = 32;
constexpr int PAD   = 3;                    // patch_size 7 -> pad 3
constexpr int HALO  = TILE + 2 * PAD;       // 38
constexpr int PITCH = 40;                   // LDS row pitch in words
constexpr int SPL   = HALO * PITCH;         // 1520 words / plane
constexpr int NPOS  = HALO * HALO;          // 1444 halo positions
constexpr int NT    = 256;                  // 8 wave32 per block

// LDS plane offsets (in words). 10 planes * 1520 * 4B = 60.8 KB.
constexpr int O_R  = 0 * SPL, O_G  = 1 * SPL, O_B  = 2 * SPL;   // raw rgb
constexpr int O_PR = 3 * SPL, O_PG = 4 * SPL, O_PB = 5 * SPL;   // raw pred rgb
constexpr int O_A  = 6 * SPL, O_C  = 7 * SPL;                   // pca ch1/ch2 of x
constexpr int O_AP = 8 * SPL, O_BP = 9 * SPL;                   // pca ch1/ch2 of pred

// Minv (inverse of M, matching the float64->float32 reference values).
constexpr float KINV = 0.50000500005f;      // 1 / (3 * 0.66666)
}

// Generic -> LDS / global addrspace pointers, int-typed to match the async
// builtin's signature ('__device__ int *' == addrspace(1) int*). The low 32
// bits of a generic LDS address are the LDS byte offset on amdgcn; the
// integer route keeps the host pass happy.
__device__ __forceinline__ AS3 int* to_lds(float* p) {
    return (AS3 int*)(uint32_t)(uint64_t)(uintptr_t)p;
}
__device__ __forceinline__ AS1 int* to_glob(const float* p) {
    return (AS1 int*)(uint64_t)(uintptr_t)p;
}

// One 4-byte element, global -> LDS. Async DMA path when available.
__device__ __forceinline__ void stage_b32(float* ldsDst, const float* gSrc) {
#ifdef HAVE_ASYNC
    __builtin_amdgcn_global_load_async_to_lds_b32(to_glob(gSrc), to_lds(ldsDst),
                                                  /*offset=*/0, /*cpol=*/0);
#else
    *ldsDst = *gSrc;
#endif
}

__device__ __forceinline__ void wait_async_done() {
#ifdef HAVE_ASYNC
#if __has_builtin(__builtin_amdgcn_s_wait_asynccnt)
    __builtin_amdgcn_s_wait_asynccnt(0);
#else
    asm volatile("s_wait_asynccnt 0" ::: "memory");
#endif
#endif
}

__global__ __launch_bounds__(NT)
void wavg_color_kernel(const float* __restrict__ xr,   // [N][3][256][256]
                       const float* __restrict__ xp,   // [N][3][256][256]
                       float* __restrict__ out) {      // [N][3][256][256]
    __shared__ float sm[10 * SPL];

    const int tid = threadIdx.x;
    const int n   = blockIdx.y;
    const int ty  = blockIdx.x >> 3;
    const int tx  = blockIdx.x & 7;
    const int y0  = ty * TILE;
    const int x0  = tx * TILE;

    const float* rgb = xr + n * 3 * PLANE;
    const float* prd = xp + n * 3 * PLANE;
    float*       outn = out + n * 3 * PLANE;

    // ---------------- Phase 1a: stage raw halo planes into LDS (async) -----
    for (int t = tid; t < NPOS; t += NT) {
        const int hy = t / HALO;
        const int hx = t - hy * HALO;
        const int gy = y0 - PAD + hy;
        const int gx = x0 - PAD + hx;
        const int li = hy * PITCH + hx;
        if (gy >= 0 && gy < Himg && gx >= 0 && gx < Wimg) {
            const int g = gy * Wimg + gx;
            stage_b32(sm + O_R  + li, rgb + 0 * PLANE + g);
            stage_b32(sm + O_G  + li, rgb + 1 * PLANE + g);
            stage_b32(sm + O_B  + li, rgb + 2 * PLANE + g);
            stage_b32(sm + O_PR + li, prd + 0 * PLANE + g);
            stage_b32(sm + O_PG + li, prd + 1 * PLANE + g);
            stage_b32(sm + O_PB + li, prd + 2 * PLANE + g);
        }
    }
    wait_async_done();
    __syncthreads();

    // ---------------- Phase 1b: rgb -> pca (ch1 = r-b, ch2 = g-(r+b)/2) ----
    // Zero-padded outside the image, matching jnp.pad on the pca planes.
    for (int t = tid; t < NPOS; t += NT) {
        const int hy = t / HALO;
        const int hx = t - hy * HALO;
        const int gy = y0 - PAD + hy;
        const int gx = x0 - PAD + hx;
        const int li = hy * PITCH + hx;
        float a = 0.f, c = 0.f, ap = 0.f, bp = 0.f;
        if (gy >= 0 && gy < Himg && gx >= 0 && gx < Wimg) {
            const float r  = sm[O_R  + li], g  = sm[O_G  + li], b  = sm[O_B  + li];
            const float pr = sm[O_PR + li], pg = sm[O_PG + li], pb = sm[O_PB + li];
            a  = r - b;
            c  = g - 0.5f * (r + b);
            ap = pr - pb;
            bp = pg - 0.5f * (pr + pb);
        }
        sm[O_A  + li] = a;
        sm[O_C  + li] = c;
        sm[O_AP + li] = ap;
        sm[O_BP + li] = bp;
    }
    __syncthreads();

    // ---------------- Phase 2: 7x7 softmax stencil + pca->rgb --------------
    const int lane = tid & 31;
    const int wave = tid >> 5;

#ifdef HAVE_WMMA
    // A-matrix (16x4 f32, documented layout: lanes 0-15 hold K0,K1; lanes
    // 16-31 hold K2,K3; M = lane&15). A[m][k] = Minv[k][m]; K=3 column holds
    // the +0.5 bias (paired with B row3 = 1.0). Rows m>=3 are zero.
    const int hi = lane >> 4;
    const int m  = lane & 15;
    v2f Av;
    {
        const float r0 = (m < 3) ? KINV : 0.f;                                   // Minv[0][m]
        const float r1 = (m == 0) ? 0.5f : ((m == 2) ? -0.5f : 0.f);             // Minv[1][m]
        const float r2 = (m == 0) ? -0.33333334f
                       : ((m == 1) ? 0.66666667f : ((m == 2) ? -0.33333334f : 0.f)); // Minv[2][m]
        const float r3 = (m < 3) ? 0.5f : 0.f;                                   // bias column
        Av.x = hi ? r2 : r0;   // VGPR0: K0 (lanes<16) / K2 (lanes>=16)
        Av.y = hi ? r3 : r1;   // VGPR1: K1 (lanes<16) / K3 (lanes>=16)
    }
#endif

    #pragma unroll
    for (int it = 0; it < 4; ++it) {
        const int r = it * 8 + wave;   // tile row 0..31 (one full row per wave)
        const int c = lane;            // tile col 0..31

        const int ci = (r + PAD) * PITCH + (c + PAD);
        const float a0 = sm[O_A + ci];
        const float b0 = sm[O_C + ci];

        float wsum = 0.f, sa = 0.f, sb = 0.f;
        for (int dy = 0; dy < 7; ++dy) {
            const int ro = (r + dy) * PITCH + c;
            #pragma unroll
            for (int dx = 0; dx < 7; ++dx) {
                const float la  = sm[O_A  + ro + dx];
                const float lb  = sm[O_C  + ro + dx];
                const float da  = la - a0;
                const float db  = lb - b0;
                const float d   = da * da + db * db;
                const float w   = __expf(-d);          // max(-d)=0 at center: safe
                wsum += w;
                sa   += w * sm[O_AP + ro + dx];
                sb   += w * sm[O_BP + ro + dx];
            }
        }
        const float inv = __builtin_amdgcn_rcpf(wsum);
        sa *= inv;
        sb *= inv;
        // pred pca channel 0 at the center (raw pred rgb is still in LDS):
        const float p0 = 0.66666f * (sm[O_PR + ci] + sm[O_PG + ci] + sm[O_PB + ci] - 1.5f);

#ifdef HAVE_WMMA
        // D(16x16) = A(16x4) x B(4x16): 16 pixels per WMMA, two per wave row.
        // B layout (4x16 f32): VGPR0 = row K0 (lanes<16) / K2 (lanes>=16),
        // VGPR1 = row K1 / K3; N = lane&15. Rows: [p0, sa, sb, 1].
        #pragma unroll
        for (int g = 0; g < 2; ++g) {
            const int srcl = (g << 4) | m;
            const float p0s = __shfl(p0, srcl, 32);
            const float sas = __shfl(sa, srcl, 32);
            const float sbs = __shfl(sb, srcl, 32);
            v2f Bv;
            Bv.x = hi ? sbs : p0s;
            Bv.y = hi ? 1.0f : sas;
            v8f Cv = {};
            v8f Dv = __builtin_amdgcn_wmma_f32_16x16x4_f32(
                false, Av, false, Bv, (short)0, Cv, false, false);
            if (lane < 16) {   // D rows 0..2 live in lanes 0-15, VGPRs 0..2
                const int o = (y0 + r) * Wimg + (x0 + (g << 4) + lane);
                outn[0 * PLANE + o] = Dv[0];
                outn[1 * PLANE + o] = Dv[1];
                outn[2 * PLANE + o] = Dv[2];
            }
        }
#else
        const float o0 = KINV * p0 + 0.5f * sa - 0.33333334f * sb + 0.5f;
        const float o1 = KINV * p0 + 0.66666667f * sb + 0.5f;
        const float o2 = KINV * p0 - 0.5f * sa - 0.33333334f * sb + 0.5f;
        const int o = (y0 + r) * Wimg + (x0 + c);
        outn[0 * PLANE + o] = o0;
        outn[1 * PLANE + o] = o1;
        outn[2 * PLANE + o] = o2;
#endif
    }
}

extern "C" void kernel_launch(void* const* d_in, const int* in_sizes, int n_in,
                              void* d_out, int out_size, void* d_ws, size_t ws_size,
                              hipStream_t stream) {
    (void)n_in; (void)out_size; (void)d_ws; (void)ws_size;
    const float* xr = (const float*)d_in[0];
    const float* xp = (const float*)d_in[1];
    float* out = (float*)d_out;
    const int N = in_sizes[0] / (3 * PLANE);   // 8 for the reference shapes
    dim3 grid((Himg / TILE) * (Wimg / TILE), N);
    wavg_color_kernel<<<grid, NT, 0, stream>>>(xr, xp, out);
}